// KAttention_19971597926623
// MI455X (gfx1250) — compile-verified
//
#include <hip/hip_runtime.h>
#include <hip/hip_bf16.h>
#include <math.h>

#define N_PTS 8192
#define C_CH  256
#define K_NN  16
#define LDS_STRIDE 272                       // 256 + 16 dwords of TDM padding
#define LDS_BYTES  (LDS_STRIDE * C_CH * 4)   // 278528 B <= 320KB/WGP

typedef float v2f __attribute__((ext_vector_type(2)));
typedef float v8f __attribute__((ext_vector_type(8)));
typedef unsigned int u32x4 __attribute__((ext_vector_type(4)));
typedef int i32x4 __attribute__((ext_vector_type(4)));
typedef int i32x8 __attribute__((ext_vector_type(8)));

// ---------------------------------------------------------------------------
// TDM: stage a 256x256 fp32 weight matrix into LDS with a 272-dword row
// stride (pad_interval = 256 dwords, pad_amount = 16 dwords) so the two
// k-rows read by opposite lane halves hit disjoint LDS banks.
// D# encoding per CDNA5 ISA 8.3-8.6.
// ---------------------------------------------------------------------------
__device__ __forceinline__ void tdm_stage_weight(const float* gptr) {
    const unsigned long long ga = (unsigned long long)(size_t)gptr;
    u32x4 g0;
    g0[0] = 1u;                                   // count=1, user D#, no gather
    g0[1] = 0u;                                   // lds_addr = 0 (dynamic LDS base)
    g0[2] = (unsigned)(ga & 0xFFFFFFFFu);         // global_addr[31:0]
    g0[3] = (unsigned)((ga >> 32) & 0x1FFFFFFu)   // global_addr[56:32]
            | (2u << 30);                         // type = 2 ("image")
    i32x8 g1;
    g1[0] = (int)0x1FF20000u;   // data_size=4B | pad_enable | interval=256dw | amount=16dw
    g1[1] = (int)(256u << 16);  // tensor_dim0[15:0] = 256
    g1[2] = (int)(256u << 16);  // tensor_dim0[31:16]=0 | tensor_dim1[15:0] = 256
    g1[3] = (int)(256u << 16);  // tensor_dim1[31:16]=0 | tile_dim0 = 256
    g1[4] = (int)256;           // tile_dim1 = 256, tile_dim2 = 0
    g1[5] = (int)256;           // tensor_dim0_stride[31:0] = 256
    g1[6] = 0;
    g1[7] = 0;
    const i32x4 z4 = {0, 0, 0, 0};
#if defined(__clang_major__) && (__clang_major__ >= 23)
    const i32x8 z8 = {0, 0, 0, 0, 0, 0, 0, 0};
    __builtin_amdgcn_tensor_load_to_lds(g0, g1, z4, z4, z8, 0);
#else
    __builtin_amdgcn_tensor_load_to_lds(g0, g1, z4, z4, 0);
#endif
    __builtin_amdgcn_s_wait_tensorcnt(0);
}

// ---------------------------------------------------------------------------
// Kernel 1: exact KNN (K=16 incl. self), one wave32 per query point.
// ---------------------------------------------------------------------------
__global__ __launch_bounds__(256) void knn_kernel(const float* __restrict__ coords,
                                                  int* __restrict__ idx_out) {
    const int lane = threadIdx.x & 31;
    const int wave = threadIdx.x >> 5;
    const int n = blockIdx.x * 8 + wave;
    if (n >= N_PTS) return;

    __builtin_prefetch(&coords[(blockIdx.x * 8) * 3], 0, 3);

    const float cx = coords[n * 3 + 0];
    const float cy = coords[n * 3 + 1];
    const float cz = coords[n * 3 + 2];

    float bd[K_NN];
    int   bi[K_NN];
#pragma unroll
    for (int s = 0; s < K_NN; ++s) { bd[s] = 3.4e38f; bi[s] = 0x7fffffff; }

    for (int j = lane; j < N_PTS; j += 32) {
        const float dx = coords[j * 3 + 0] - cx;
        const float dy = coords[j * 3 + 1] - cy;
        const float dz = coords[j * 3 + 2] - cz;
        const float d = dx * dx + dy * dy + dz * dz;
        if (d < bd[K_NN - 1] || (d == bd[K_NN - 1] && j < bi[K_NN - 1])) {
            float td = d; int ti = j;
#pragma unroll
            for (int s = 0; s < K_NN; ++s) {
                if (td < bd[s] || (td == bd[s] && ti < bi[s])) {
                    float t1 = bd[s]; bd[s] = td; td = t1;
                    int   t2 = bi[s]; bi[s] = ti; ti = t2;
                }
            }
        }
    }

    // Merge 32 sorted lists: 16 rounds of wave-wide argmin over the heads.
    for (int r = 0; r < K_NN; ++r) {
        const float md = bd[0];
        const int   mi = bi[0];
        float rd = md; int ri = mi;
#pragma unroll
        for (int off = 16; off > 0; off >>= 1) {
            const float od = __shfl_xor(rd, off, 32);
            const int   oi = __shfl_xor(ri, off, 32);
            if (od < rd || (od == rd && oi < ri)) { rd = od; ri = oi; }
        }
        if (lane == 0) idx_out[n * K_NN + r] = ri;
        if (ri == mi) {
#pragma unroll
            for (int s = 0; s < K_NN - 1; ++s) { bd[s] = bd[s + 1]; bi[s] = bi[s + 1]; }
            bd[K_NN - 1] = 3.4e38f; bi[K_NN - 1] = 0x7fffffff;
        }
    }
}

// ---------------------------------------------------------------------------
// Kernel 2: out[M,C] = A[M,C] @ B[C,C] + bias (+ residual), fp32 WMMA.
// B is TDM-staged into LDS (272-dword padded rows); block = 8 waves, wave
// computes a 16x64 tile, K stepped by 4 with V_WMMA_F32_16X16X4_F32.
// Fragments (ISA 7.12.2): A/B lane l -> row/col (l&15), VGPRs = K pair
// {2*(l>>4), 2*(l>>4)+1}; D lane l -> col (l&15), VGPR r = row r+8*(l>>4).
// ---------------------------------------------------------------------------
template <bool HAS_RES>
__global__ __launch_bounds__(256) void gemm_wmma(const float* __restrict__ A,
                                                 const float* __restrict__ B,
                                                 const float* __restrict__ bias,
                                                 const float* __restrict__ residual,
                                                 float* __restrict__ out) {
    extern __shared__ float sB[];  // [256][272]
    const int lane = threadIdx.x & 31;
    const int wave = threadIdx.x >> 5;
    const int hi = lane >> 4;
    const int ln = lane & 15;
    const int m0 = blockIdx.x * 32 + (wave >> 2) * 16;
    const int n0base = (wave & 3) * 64;

    if (wave == 0) tdm_stage_weight(B);
    __syncthreads();

    v8f acc[4];
#pragma unroll
    for (int t = 0; t < 4; ++t) acc[t] = (v8f){};

    const int arow = m0 + ln;
    for (int k0 = 0; k0 < C_CH; k0 += 4) {
        const int kk = k0 + hi * 2;
        const v2f a = *(const v2f*)&A[arow * C_CH + kk];
#pragma unroll
        for (int t = 0; t < 4; ++t) {
            const int c = n0base + t * 16 + ln;
            v2f b;
            b.x = sB[kk * LDS_STRIDE + c];
            b.y = sB[(kk + 1) * LDS_STRIDE + c];
            acc[t] = __builtin_amdgcn_wmma_f32_16x16x4_f32(
                false, a, false, b, (short)0, acc[t], false, false);
        }
    }

#pragma unroll
    for (int t = 0; t < 4; ++t) {
        const int c = n0base + t * 16 + ln;
        const float bv = bias[c];
#pragma unroll
        for (int r = 0; r < 8; ++r) {
            const int m = m0 + r + 8 * hi;
            float v = acc[t][r] + bv;
            if (HAS_RES) v += residual[m * C_CH + c];
            out[m * C_CH + c] = v;
        }
    }
}

// ---------------------------------------------------------------------------
// Kernel 3: fused pe-MLP + delta-features + softmax(K) + weighted g-sum.
// One wave32 per point; 8 points/block share one TDM-staged pe1_w2 in LDS.
// A-tile = the point's 16 neighbors (M=16), hidden activations recomputed
// per k-step from delta-coords; 16 output channel tiles live in registers.
// ---------------------------------------------------------------------------
__global__ __launch_bounds__(256) void attn_kernel(const float* __restrict__ coords,
                                                   const int* __restrict__ nn_idx,
                                                   const float* __restrict__ theta_f,
                                                   const float* __restrict__ phi_f,
                                                   const float* __restrict__ g_f,
                                                   const float* __restrict__ w1,
                                                   const float* __restrict__ b1,
                                                   const float* __restrict__ w2,
                                                   const float* __restrict__ b2,
                                                   float* __restrict__ y) {
    extern __shared__ float sW2[];  // [256][272]
    const int lane = threadIdx.x & 31;
    const int wave = threadIdx.x >> 5;
    const int n = blockIdx.x * 8 + wave;
    const int hi = lane >> 4;
    const int ln = lane & 15;

    if (wave == 0) tdm_stage_weight(w2);
    __syncthreads();

    // Neighbor (row of A) this lane feeds: slot ln; lanes l and l+16 share it.
    const int jn = nn_idx[n * K_NN + ln];
    const float dx = coords[jn * 3 + 0] - coords[n * 3 + 0];
    const float dy = coords[jn * 3 + 1] - coords[n * 3 + 1];
    const float dz = coords[jn * 3 + 2] - coords[n * 3 + 2];

    v8f acc[16];
#pragma unroll
    for (int t = 0; t < 16; ++t) acc[t] = (v8f){};

    for (int k0 = 0; k0 < C_CH; k0 += 4) {
        const int kk = k0 + hi * 2;
        // hidden = relu(delta @ w1 + b1): two contraction channels per lane
        v2f a;
        {
            const float h0 = dx * w1[0 * C_CH + kk] + dy * w1[1 * C_CH + kk] +
                             dz * w1[2 * C_CH + kk] + b1[kk];
            const float h1 = dx * w1[0 * C_CH + kk + 1] + dy * w1[1 * C_CH + kk + 1] +
                             dz * w1[2 * C_CH + kk + 1] + b1[kk + 1];
            a.x = fmaxf(h0, 0.0f);
            a.y = fmaxf(h1, 0.0f);
        }
#pragma unroll
        for (int t = 0; t < 16; ++t) {
            const int c = t * 16 + ln;
            v2f b;
            b.x = sW2[kk * LDS_STRIDE + c];
            b.y = sW2[(kk + 1) * LDS_STRIDE + c];
            acc[t] = __builtin_amdgcn_wmma_f32_16x16x4_f32(
                false, a, false, b, (short)0, acc[t], false, false);
        }
    }

    // Epilogue: lane holds pe[nbr r+8*hi][channel t*16+ln] for r=0..7.
    int nbr[8];
#pragma unroll
    for (int r = 0; r < 8; ++r) nbr[r] = nn_idx[n * K_NN + 8 * hi + r];

#pragma unroll
    for (int t = 0; t < 16; ++t) {
        const int c = t * 16 + ln;
        const float th = theta_f[n * C_CH + c];
        const float bb = b2[c];
        float e[8];
        float m = -3.4e38f;
#pragma unroll
        for (int r = 0; r < 8; ++r) {
            const float pe = acc[t][r] + bb;
            const float df = (pe * (th - phi_f[nbr[r] * C_CH + c]) + pe) * 0.0625f;
            e[r] = df;
            m = fmaxf(m, df);
        }
        m = fmaxf(m, __shfl_xor(m, 16, 32));  // partner lane holds the other 8
        float s = 0.0f, yv = 0.0f;
#pragma unroll
        for (int r = 0; r < 8; ++r) {
            const float ex = __expf(e[r] - m);
            s += ex;
            yv += ex * g_f[nbr[r] * C_CH + c];
        }
        s  += __shfl_xor(s, 16, 32);
        yv += __shfl_xor(yv, 16, 32);
        if (hi == 0) y[n * C_CH + c] = yv / s;
    }
}

// ---------------------------------------------------------------------------
extern "C" void kernel_launch(void* const* d_in, const int* in_sizes, int n_in,
                              void* d_out, int out_size, void* d_ws, size_t ws_size,
                              hipStream_t stream) {
    const float* coords  = (const float*)d_in[0];
    const float* feats   = (const float*)d_in[1];
    const float* theta_w = (const float*)d_in[2];
    const float* theta_b = (const float*)d_in[3];
    const float* phi_w   = (const float*)d_in[4];
    const float* phi_b   = (const float*)d_in[5];
    const float* g_w     = (const float*)d_in[6];
    const float* g_b     = (const float*)d_in[7];
    const float* pe1_w1  = (const float*)d_in[8];
    const float* pe1_b1  = (const float*)d_in[9];
    const float* pe1_w2  = (const float*)d_in[10];
    const float* pe1_b2  = (const float*)d_in[11];
    const float* W_w     = (const float*)d_in[12];
    const float* W_b     = (const float*)d_in[13];
    float* out = (float*)d_out;

    // Workspace carve-up
    int*   nn_idx  = (int*)d_ws;
    float* theta_f = (float*)((char*)d_ws + (size_t)N_PTS * K_NN * sizeof(int));
    float* phi_f   = theta_f + (size_t)N_PTS * C_CH;
    float* g_f     = phi_f   + (size_t)N_PTS * C_CH;
    float* y       = g_f     + (size_t)N_PTS * C_CH;

    // Allow >64KB dynamic LDS (ignore failures; AMD devices generally permit)
    (void)hipFuncSetAttribute((const void*)gemm_wmma<false>,
                              hipFuncAttributeMaxDynamicSharedMemorySize, LDS_BYTES);
    (void)hipFuncSetAttribute((const void*)gemm_wmma<true>,
                              hipFuncAttributeMaxDynamicSharedMemorySize, LDS_BYTES);
    (void)hipFuncSetAttribute((const void*)attn_kernel,
                              hipFuncAttributeMaxDynamicSharedMemorySize, LDS_BYTES);

    // 1) KNN indices
    knn_kernel<<<N_PTS / 8, 256, 0, stream>>>(coords, nn_idx);

    // 2) Per-point projections (gather commutes with the linear layers)
    gemm_wmma<false><<<N_PTS / 32, 256, LDS_BYTES, stream>>>(feats, theta_w, theta_b, nullptr, theta_f);
    gemm_wmma<false><<<N_PTS / 32, 256, LDS_BYTES, stream>>>(feats, phi_w,   phi_b,   nullptr, phi_f);
    gemm_wmma<false><<<N_PTS / 32, 256, LDS_BYTES, stream>>>(feats, g_w,     g_b,     nullptr, g_f);

    // 3) Fused pe-MLP + attention + aggregation
    attn_kernel<<<N_PTS / 8, 256, LDS_BYTES, stream>>>(coords, nn_idx, theta_f, phi_f, g_f,
                                                       pe1_w1, pe1_b1, pe1_w2, pe1_b2, y);

    // 4) Output projection + residual
    gemm_wmma<true><<<N_PTS / 32, 256, LDS_BYTES, stream>>>(y, W_w, W_b, feats, out);
}